// QFNN_14963666059661
// MI455X (gfx1250) — compile-verified
//
#include <hip/hip_runtime.h>
#include <hip/hip_bf16.h>

#define VOCAB   32000
#define HIDDEN  768
#define NLAYERS 4
#define NTOK    2048
#define BATCH   2
#define ROWS    (BATCH * NTOK)   // 4096

typedef float v2f __attribute__((ext_vector_type(2)));
typedef float v8f __attribute__((ext_vector_type(8)));
typedef int   v4i __attribute__((ext_vector_type(4)));

#if __has_builtin(__builtin_amdgcn_global_load_async_to_lds_b128) && \
    __has_builtin(__builtin_amdgcn_s_wait_asynccnt)
#define USE_ASYNC_LDS 1
#else
#define USE_ASYNC_LDS 0
#endif

typedef __attribute__((address_space(1))) v4i* g_v4i;   // global int4*
typedef __attribute__((address_space(3))) v4i* l_v4i;   // LDS int4*

// ---------------------------------------------------------------- embedding
__global__ void embed_kernel(const int* __restrict__ ids,
                             const float* __restrict__ emb,
                             float* __restrict__ states) {
  int t = blockIdx.x * blockDim.x + threadIdx.x;
  if (t < ROWS) {
    int id = ids[t];
    states[2 * t + 0] = emb[2 * id + 0];
    states[2 * t + 1] = emb[2 * id + 1];
  }
}

// --------------------------------------------------------------- attention
__device__ __forceinline__ void attn_row(const float2* __restrict__ sS,
                                         float xi, float yi, int lane,
                                         float& cx, float& cy, float& rs) {
  cx = 0.f; cy = 0.f; rs = 0.f;
  for (int j = lane; j < NTOK; j += 32) {
    float2 s = sS[j];
    float sim = xi * s.x + yi * s.y + 0.1f * (xi * s.y - yi * s.x);
    float den = 1e-6f + fabsf(sim);
    float attn = (den < 2.0f) ? (sim / den) : 0.f;
    cx += attn * s.x;
    cy += attn * s.y;
    rs += attn;
  }
  for (int off = 16; off > 0; off >>= 1) {
    cx += __shfl_down(cx, off, 32);
    cy += __shfl_down(cy, off, 32);
    rs += __shfl_down(rs, off, 32);
  }
}

__global__ __launch_bounds__(256)
void attn_pass1(const float* __restrict__ states, float* __restrict__ k1,
                float* __restrict__ mid, float* __restrict__ rowsum) {
  __shared__ float2 sS[NTOK];
  const int lane = threadIdx.x & 31;
  const int wave = threadIdx.x >> 5;
  const int row  = blockIdx.x * 8 + wave;     // [0, 4096)
  const int base = (row / NTOK) * NTOK;
  for (int j = threadIdx.x; j < NTOK; j += blockDim.x) {
    sS[j] = make_float2(states[2 * (base + j)], states[2 * (base + j) + 1]);
  }
  __syncthreads();
  const float xi = states[2 * row], yi = states[2 * row + 1];
  float cx, cy, rs;
  attn_row(sS, xi, yi, lane, cx, cy, rs);
  if (lane == 0) {
    float k1x = 0.1f * cx, k1y = 0.1f * cy;     // DT * ctx
    k1[2 * row]      = k1x;  k1[2 * row + 1] = k1y;
    mid[2 * row]     = xi + k1x;
    mid[2 * row + 1] = yi + k1y;
    rowsum[row] = rs;
  }
}

__global__ __launch_bounds__(256)
void attn_pass2(float* __restrict__ states, const float* __restrict__ k1,
                const float* __restrict__ mid, const float* __restrict__ rowsum) {
  __shared__ float2 sS[NTOK];
  const int lane = threadIdx.x & 31;
  const int wave = threadIdx.x >> 5;
  const int row  = blockIdx.x * 8 + wave;
  const int base = (row / NTOK) * NTOK;
  for (int j = threadIdx.x; j < NTOK; j += blockDim.x) {
    sS[j] = make_float2(mid[2 * (base + j)], mid[2 * (base + j) + 1]);
  }
  __syncthreads();
  const float mx = mid[2 * row], my = mid[2 * row + 1];
  float cx, cy, rs;
  attn_row(sS, mx, my, lane, cx, cy, rs);
  if (lane == 0) {
    float k2x = 0.1f * cx, k2y = 0.1f * cy;
    float sx = states[2 * row], sy = states[2 * row + 1];
    float ex = sx + 0.5f * (k1[2 * row]     + k2x);
    float ey = sy + 0.5f * (k1[2 * row + 1] + k2y);
    float sw = 1.f / (1.f + expf(0.9f));        // sigmoid(BETA*(DT-1))
    float nx = sw * sx + (1.f - sw) * ex;
    float ny = sw * sy + (1.f - sw) * ey;
    float r  = sqrtf(nx * nx + ny * ny);
    float nr = fminf(fmaxf(r + rowsum[row] * 0.01f, 0.1f), 2.0f);
    float sc = nr / (r + 1e-8f);
    states[2 * row]     = nx * sc;
    states[2 * row + 1] = ny * sc;
  }
}

// ------------------------------------------------------------- projection
__global__ void hidden_kernel(const float* __restrict__ states,
                              const float* __restrict__ pw,
                              const float* __restrict__ pb,
                              float* __restrict__ hidden) {
  int t = blockIdx.x * blockDim.x + threadIdx.x;
  if (t < ROWS * HIDDEN) {
    int r = t / HIDDEN, h = t - r * HIDDEN;
    hidden[t] = states[2 * r] * pw[h] + states[2 * r + 1] * pw[HIDDEN + h] + pb[h];
  }
}

// ----------------------------------------------------------- WMMA fp32 GEMM
// C[4096 x 32000] = hidden[4096 x 768] @ out_w[768 x 32000] + out_b
#define BM  64
#define BN  256
#define KC  32
#define NIT (HIDDEN / KC)   // 24 K-tiles
#define LDA 36     // floats; 144B row stride: 16B-aligned, conflict-free frags
#define LDB 272    // floats; 1088B row stride: 16B-aligned, conflict-free frags

// Stage one 64x32 A tile + 32x256 B tile into LDS (10 x b128 per thread).
__device__ __forceinline__ void stage_tiles(const float* __restrict__ A,
                                            const float* __restrict__ Bm,
                                            float* bufA, float* bufB,
                                            int row0, int col0, int k0, int tid) {
#if USE_ASYNC_LDS
#pragma unroll
  for (int r = 0; r < 2; r++) {
    int lin = tid + r * 256;
    int arow = lin >> 3;
    int ac4  = (lin & 7) << 2;
    __builtin_amdgcn_global_load_async_to_lds_b128(
        (g_v4i)(A + (row0 + arow) * HIDDEN + k0 + ac4),
        (l_v4i)(bufA + arow * LDA + ac4), 0, 0);
  }
#pragma unroll
  for (int r = 0; r < 8; r++) {
    int lin = tid + r * 256;
    int brow = lin >> 6;
    int bc4  = (lin & 63) << 2;
    __builtin_amdgcn_global_load_async_to_lds_b128(
        (g_v4i)(Bm + (k0 + brow) * VOCAB + col0 + bc4),
        (l_v4i)(bufB + brow * LDB + bc4), 0, 0);
  }
#else
#pragma unroll
  for (int r = 0; r < 2; r++) {
    int lin = tid + r * 256;
    int arow = lin >> 3;
    int ac4  = (lin & 7) << 2;
    float4 v = *(const float4*)&A[(row0 + arow) * HIDDEN + k0 + ac4];
    *(float4*)&bufA[arow * LDA + ac4] = v;
  }
#pragma unroll
  for (int r = 0; r < 8; r++) {
    int lin = tid + r * 256;
    int brow = lin >> 6;
    int bc4  = (lin & 63) << 2;
    float4 v = *(const float4*)&Bm[(k0 + brow) * VOCAB + col0 + bc4];
    *(float4*)&bufB[brow * LDB + bc4] = v;
  }
#endif
}

__global__ __launch_bounds__(256)
void gemm_wmma(const float* __restrict__ A, const float* __restrict__ Bm,
               const float* __restrict__ bias, float* __restrict__ C) {
  __shared__ __align__(16) float sA[2][BM * LDA];   // 2 x  9.0 KB
  __shared__ __align__(16) float sB[2][KC * LDB];   // 2 x 34.0 KB
  const int tid  = threadIdx.x;
  const int lane = tid & 31;
  const int wave = tid >> 5;
  const int mw = wave >> 2;          // 0..1 : 32-row slab
  const int nw = wave & 3;           // 0..3 : 64-col slab
  const int lhalf = lane >> 4;       // 0/1  : K-pair select per ISA layout
  const int l15   = lane & 15;
  const int col0 = blockIdx.x * BN;
  const int row0 = blockIdx.y * BM;

  v8f acc[2][4];
#pragma unroll
  for (int i = 0; i < 2; i++)
#pragma unroll
    for (int j = 0; j < 4; j++) acc[i][j] = {};

  // prologue: stage K-tile 0 into buffer 0
  stage_tiles(A, Bm, &sA[0][0], &sB[0][0], row0, col0, 0, tid);

  for (int it = 0; it < NIT; ++it) {
    const int cur = it & 1;
    // prefetch next K-tile into the other buffer, then wait for current tile
    if (it + 1 < NIT) {
      stage_tiles(A, Bm, &sA[cur ^ 1][0], &sB[cur ^ 1][0],
                  row0, col0, (it + 1) * KC, tid);
#if USE_ASYNC_LDS
      __builtin_amdgcn_s_wait_asynccnt(10);   // only the 10 just-issued remain
#endif
    } else {
#if USE_ASYNC_LDS
      __builtin_amdgcn_s_wait_asynccnt(0);
#endif
    }
    __syncthreads();   // current tile visible to all waves

    const float* __restrict__ cA = &sA[cur][0];
    const float* __restrict__ cB = &sB[cur][0];
#pragma unroll
    for (int kk = 0; kk < KC; kk += 4) {
      const int c = kk + lhalf * 2;          // even -> b64-aligned frag loads
      v2f a[2];
#pragma unroll
      for (int ms = 0; ms < 2; ms++) {
        int arow = mw * 32 + ms * 16 + l15;
        float2 av = *(const float2*)&cA[arow * LDA + c];
        a[ms].x = av.x; a[ms].y = av.y;
      }
#pragma unroll
      for (int ns = 0; ns < 4; ns++) {
        int n = nw * 64 + ns * 16 + l15;
        v2f b;
        b.x = cB[c * LDB + n];
        b.y = cB[(c + 1) * LDB + n];
#pragma unroll
        for (int ms = 0; ms < 2; ms++) {
          acc[ms][ns] = __builtin_amdgcn_wmma_f32_16x16x4_f32(
              false, a[ms], false, b, (short)0, acc[ms][ns], false, false);
        }
      }
    }
    __syncthreads();   // all waves done reading buf[cur] before it is reloaded
  }

  // epilogue: bias + store (C/D layout: VGPR r -> M = r + 8*lhalf, N = l15)
#pragma unroll
  for (int ns = 0; ns < 4; ns++) {
    int col = col0 + nw * 64 + ns * 16 + l15;
    float bv = bias[col];
#pragma unroll
    for (int ms = 0; ms < 2; ms++) {
      int rbase = row0 + mw * 32 + ms * 16 + lhalf * 8;
#pragma unroll
      for (int r = 0; r < 8; r++) {
        C[(rbase + r) * (long)VOCAB + col] = acc[ms][ns][r] + bv;
      }
    }
  }
}

// ------------------------------------------------------------------ launch
extern "C" void kernel_launch(void* const* d_in, const int* in_sizes, int n_in,
                              void* d_out, int out_size, void* d_ws, size_t ws_size,
                              hipStream_t stream) {
  (void)in_sizes; (void)n_in; (void)out_size; (void)ws_size;
  const int*   ids = (const int*)d_in[0];
  const float* emb = (const float*)d_in[1];
  const float* pw  = (const float*)d_in[2];
  const float* pb  = (const float*)d_in[3];
  const float* ow  = (const float*)d_in[4];
  const float* ob  = (const float*)d_in[5];
  float* out = (float*)d_out;

  float* ws     = (float*)d_ws;
  float* states = ws;                 // 8192 floats
  float* k1     = ws + 8192;          // 8192
  float* mid    = ws + 16384;         // 8192
  float* rowsum = ws + 24576;         // 4096
  float* hidden = ws + 28672;         // 4096*768 = 3,145,728

  embed_kernel<<<(ROWS + 255) / 256, 256, 0, stream>>>(ids, emb, states);
  for (int l = 0; l < NLAYERS; l++) {
    attn_pass1<<<ROWS / 8, 256, 0, stream>>>(states, k1, mid, rowsum);
    attn_pass2<<<ROWS / 8, 256, 0, stream>>>(states, k1, mid, rowsum);
  }
  hidden_kernel<<<(ROWS * HIDDEN + 255) / 256, 256, 0, stream>>>(states, pw, pb, hidden);
  gemm_wmma<<<dim3(VOCAB / BN, ROWS / BM), 256, 0, stream>>>(hidden, ow, ob, out);
}